// GraphSAGE_75282186764725
// MI455X (gfx1250) — compile-verified
//
#include <hip/hip_runtime.h>
#include <hip/hip_bf16.h>

// GraphSAGE 2-layer forward for MI455X (gfx1250, wave32).
// GEMMs use native f32 WMMA (V_WMMA_F32_16X16X4_F32) for exact f32 semantics;
// aggregation uses coalesced float4 gathers + global f32 atomics (memory-bound
// phase; the 192MB L2 absorbs GEMM operand reuse).

typedef float v2f __attribute__((ext_vector_type(2)));
typedef float v8f __attribute__((ext_vector_type(8)));

#define N_SRC0 500000
#define N_DST0 50000
#define N_DST1 10000
#define IN_F   256
#define HID    256
#define NCLS   64

__device__ __forceinline__ v8f wmma_f32(v2f a, v2f b, v8f c) {
    // (neg_a, A, neg_b, B, c_mod, C, reuse_a, reuse_b)
    return __builtin_amdgcn_wmma_f32_16x16x4_f32(
        false, a, false, b, (short)0, c, false, false);
}

// ---------------------------------------------------------------------------
// Edge scatter: 64 lanes per edge, each lane moves a float4 (16B) of the
// 256-float source row into the destination accumulator via f32 atomics.
// ---------------------------------------------------------------------------
__global__ void sage_scatter(const float* __restrict__ feat,
                             const int* __restrict__ src,
                             const int* __restrict__ dst,
                             float* __restrict__ msg,
                             float* __restrict__ deg,
                             int nedges) {
    int t = blockIdx.x * blockDim.x + threadIdx.x;
    int e = t >> 6;
    if (e >= nedges) return;
    int li = t & 63;
    int s = src[e];
    int d = dst[e];
    const float4 v = ((const float4*)(feat + (size_t)s * 256))[li];
    float* o = msg + (size_t)d * 256 + (size_t)li * 4;
    atomicAdd(o + 0, v.x);
    atomicAdd(o + 1, v.y);
    atomicAdd(o + 2, v.z);
    atomicAdd(o + 3, v.w);
    if (li == 0) atomicAdd(deg + d, 1.0f);
}

// msg[r][*] /= max(deg[r], 1)   (float4 per thread)
__global__ void sage_normalize(float* __restrict__ msg,
                               const float* __restrict__ deg,
                               int rows) {
    int t = blockIdx.x * blockDim.x + threadIdx.x;
    if (t >= rows * 64) return;
    int r = t >> 6;
    float s = 1.0f / fmaxf(deg[r], 1.0f);
    float4* p = (float4*)msg + t;
    float4 v = *p;
    v.x *= s; v.y *= s; v.z *= s; v.w *= s;
    *p = v;
}

// ---------------------------------------------------------------------------
// Layer 0 GEMM: H0 = relu([feat[:50000] | hneigh] @ [W0_self; W0_neigh])
// One wave per 16x64 output tile. K = 512 (256 self + 256 neighbor).
// A frag (16x4 f32): lanes 0-15 -> K={kb,kb+1}, lanes 16-31 -> K={kb+2,kb+3}.
// B frag (4x16 f32): mirrored K split; N = lane & 15.
// ---------------------------------------------------------------------------
__global__ void __launch_bounds__(32)
sage_gemm0(const float* __restrict__ feat,
           const float* __restrict__ hneigh,
           const float* __restrict__ Wself,
           const float* __restrict__ Wneigh,
           float* __restrict__ H0) {
    const int lane  = threadIdx.x;
    const int mrow  = blockIdx.x * 16 + (lane & 15);
    const int koff  = (lane >> 4) << 1;          // 0 or 2
    const int nbase = blockIdx.y * 64;
    const int ncol  = nbase + (lane & 15);

    v8f acc0 = {}, acc1 = {}, acc2 = {}, acc3 = {};

    const float* Aself = feat   + (size_t)mrow * 256;
    const float* Anei  = hneigh + (size_t)mrow * 256;

    // Phase 1: self term (K = 0..255 of the fused K=512)
    #pragma unroll 4
    for (int kb = 0; kb < 256; kb += 4) {
        const int kk = kb + koff;
        v2f a; a.x = Aself[kk]; a.y = Aself[kk + 1];
        const float* w0 = Wself + (size_t)kk * 256 + ncol;
        const float* w1 = w0 + 256;
        v2f b0 = {w0[0],  w1[0]};
        v2f b1 = {w0[16], w1[16]};
        v2f b2 = {w0[32], w1[32]};
        v2f b3 = {w0[48], w1[48]};
        acc0 = wmma_f32(a, b0, acc0);
        acc1 = wmma_f32(a, b1, acc1);
        acc2 = wmma_f32(a, b2, acc2);
        acc3 = wmma_f32(a, b3, acc3);
    }
    // Phase 2: neighbor term (K = 256..511)
    #pragma unroll 4
    for (int kb = 0; kb < 256; kb += 4) {
        const int kk = kb + koff;
        v2f a; a.x = Anei[kk]; a.y = Anei[kk + 1];
        const float* w0 = Wneigh + (size_t)kk * 256 + ncol;
        const float* w1 = w0 + 256;
        v2f b0 = {w0[0],  w1[0]};
        v2f b1 = {w0[16], w1[16]};
        v2f b2 = {w0[32], w1[32]};
        v2f b3 = {w0[48], w1[48]};
        acc0 = wmma_f32(a, b0, acc0);
        acc1 = wmma_f32(a, b1, acc1);
        acc2 = wmma_f32(a, b2, acc2);
        acc3 = wmma_f32(a, b3, acc3);
    }

    // C/D layout: VGPR r holds M = r (lanes 0-15) / M = 8+r (lanes 16-31)
    const int rbase = blockIdx.x * 16 + ((lane >> 4) << 3);
    const int cc    = nbase + (lane & 15);
    #pragma unroll
    for (int r = 0; r < 8; ++r) {
        float* o = H0 + (size_t)(rbase + r) * 256 + cc;
        o[0]  = fmaxf(acc0[r], 0.0f);
        o[16] = fmaxf(acc1[r], 0.0f);
        o[32] = fmaxf(acc2[r], 0.0f);
        o[48] = fmaxf(acc3[r], 0.0f);
    }
}

// ---------------------------------------------------------------------------
// Layer 1 GEMM: out = [H0[:10000] | hneigh1] @ [W1_self; W1_neigh]
// One wave per 16x64 tile (N = 64 total, no relu).
// ---------------------------------------------------------------------------
__global__ void __launch_bounds__(32)
sage_gemm1(const float* __restrict__ H0,
           const float* __restrict__ hneigh,
           const float* __restrict__ Wself,
           const float* __restrict__ Wneigh,
           float* __restrict__ out) {
    const int lane = threadIdx.x;
    const int mrow = blockIdx.x * 16 + (lane & 15);
    const int koff = (lane >> 4) << 1;
    const int ncol = lane & 15;

    v8f acc0 = {}, acc1 = {}, acc2 = {}, acc3 = {};

    const float* Aself = H0     + (size_t)mrow * 256;
    const float* Anei  = hneigh + (size_t)mrow * 256;

    #pragma unroll 4
    for (int kb = 0; kb < 256; kb += 4) {
        const int kk = kb + koff;
        v2f a; a.x = Aself[kk]; a.y = Aself[kk + 1];
        const float* w0 = Wself + (size_t)kk * 64 + ncol;
        const float* w1 = w0 + 64;
        v2f b0 = {w0[0],  w1[0]};
        v2f b1 = {w0[16], w1[16]};
        v2f b2 = {w0[32], w1[32]};
        v2f b3 = {w0[48], w1[48]};
        acc0 = wmma_f32(a, b0, acc0);
        acc1 = wmma_f32(a, b1, acc1);
        acc2 = wmma_f32(a, b2, acc2);
        acc3 = wmma_f32(a, b3, acc3);
    }
    #pragma unroll 4
    for (int kb = 0; kb < 256; kb += 4) {
        const int kk = kb + koff;
        v2f a; a.x = Anei[kk]; a.y = Anei[kk + 1];
        const float* w0 = Wneigh + (size_t)kk * 64 + ncol;
        const float* w1 = w0 + 64;
        v2f b0 = {w0[0],  w1[0]};
        v2f b1 = {w0[16], w1[16]};
        v2f b2 = {w0[32], w1[32]};
        v2f b3 = {w0[48], w1[48]};
        acc0 = wmma_f32(a, b0, acc0);
        acc1 = wmma_f32(a, b1, acc1);
        acc2 = wmma_f32(a, b2, acc2);
        acc3 = wmma_f32(a, b3, acc3);
    }

    const int rbase = blockIdx.x * 16 + ((lane >> 4) << 3);
    const int cc    = lane & 15;
    #pragma unroll
    for (int r = 0; r < 8; ++r) {
        float* o = out + (size_t)(rbase + r) * 64 + cc;
        o[0]  = acc0[r];
        o[16] = acc1[r];
        o[32] = acc2[r];
        o[48] = acc3[r];
    }
}

// ---------------------------------------------------------------------------
// Workspace layout (floats):
//   msg0   : 50000*256 = 12,800,000
//   deg0   :             50,000
//   msg1   : 10000*256 =  2,560,000
//   deg1   :             10,000
//   H0     : 50000*256 = 12,800,000
// Total 28,220,000 floats = ~113 MB. msg0..deg1 are contiguous -> one memset.
// ---------------------------------------------------------------------------
extern "C" void kernel_launch(void* const* d_in, const int* in_sizes, int n_in,
                              void* d_out, int out_size, void* d_ws, size_t ws_size,
                              hipStream_t stream) {
    const float* feat   = (const float*)d_in[0];
    const int*   src0   = (const int*)d_in[1];
    const int*   dst0   = (const int*)d_in[2];
    const int*   src1   = (const int*)d_in[3];
    const int*   dst1   = (const int*)d_in[4];
    const float* W0self = (const float*)d_in[5];
    const float* W0nei  = (const float*)d_in[6];
    const float* W1self = (const float*)d_in[7];
    const float* W1nei  = (const float*)d_in[8];
    float*       out    = (float*)d_out;

    const int E0 = in_sizes[1];
    const int E1 = in_sizes[3];

    float* ws   = (float*)d_ws;
    float* msg0 = ws;
    float* deg0 = msg0 + (size_t)N_DST0 * HID;
    float* msg1 = deg0 + N_DST0;
    float* deg1 = msg1 + (size_t)N_DST1 * HID;
    float* H0   = deg1 + N_DST1;

    // Zero all accumulators in one shot (contiguous region).
    const size_t zero_bytes =
        ((size_t)N_DST0 * HID + N_DST0 + (size_t)N_DST1 * HID + N_DST1) * sizeof(float);
    hipMemsetAsync(d_ws, 0, zero_bytes, stream);

    // Layer 0 aggregation
    {
        int threads = 256;
        int blocks = (int)(((size_t)E0 * 64 + threads - 1) / threads);
        sage_scatter<<<blocks, threads, 0, stream>>>(feat, src0, dst0, msg0, deg0, E0);
        int nblocks = (N_DST0 * 64 + threads - 1) / threads;
        sage_normalize<<<nblocks, threads, 0, stream>>>(msg0, deg0, N_DST0);
    }

    // Layer 0 GEMM + ReLU -> H0 (50000 x 256)
    sage_gemm0<<<dim3(N_DST0 / 16, HID / 64), 32, 0, stream>>>(
        feat, msg0, W0self, W0nei, H0);

    // Layer 1 aggregation (sources are H0 rows)
    {
        int threads = 256;
        int blocks = (int)(((size_t)E1 * 64 + threads - 1) / threads);
        sage_scatter<<<blocks, threads, 0, stream>>>(H0, src1, dst1, msg1, deg1, E1);
        int nblocks = (N_DST1 * 64 + threads - 1) / threads;
        sage_normalize<<<nblocks, threads, 0, stream>>>(msg1, deg1, N_DST1);
    }

    // Layer 1 GEMM -> out (10000 x 64)
    sage_gemm1<<<N_DST1 / 16, 32, 0, stream>>>(H0, msg1, W1self, W1nei, out);
}